// GAT_64055142252964
// MI455X (gfx1250) — compile-verified
//
#include <hip/hip_runtime.h>
#include <hip/hip_bf16.h>

typedef float v2f __attribute__((ext_vector_type(2)));
typedef float v8f __attribute__((ext_vector_type(8)));

#define HEADS 4
#define HC 128
#define NEG_SLOPE 0.2f

// ---------- helpers ----------
__device__ __forceinline__ unsigned enc_f32(float f) {
    unsigned b = __float_as_uint(f);
    return b ^ (((int)b >> 31) | 0x80000000u);   // order-preserving float->uint
}
__device__ __forceinline__ float dec_f32(unsigned u) {
    unsigned b = (u & 0x80000000u) ? (u ^ 0x80000000u) : ~u;
    return __uint_as_float(b);
}

// ---------- edge-weight sum (for self-loop mean) ----------
__global__ void ew_sum_kernel(const float* __restrict__ ew, float* __restrict__ out_sum, int E) {
    __shared__ float sm[8];
    float s = 0.f;
    for (int i = blockIdx.x * blockDim.x + threadIdx.x; i < E; i += gridDim.x * blockDim.x)
        s += ew[i];
    #pragma unroll
    for (int off = 16; off; off >>= 1) s += __shfl_down(s, off, 32);
    int lane = threadIdx.x & 31, w = threadIdx.x >> 5;
    if (lane == 0) sm[w] = s;
    __syncthreads();
    if (threadIdx.x == 0) {
        float t = 0.f;
        for (int i = 0; i < (int)(blockDim.x >> 5); ++i) t += sm[i];
        atomicAdd(out_sum, t);
    }
}

// ---------- c_h = sum_d W_e[h*32+d] * att_e[h][d]  (128 threads, wave w == head w) ----------
__global__ void edge_coef_kernel(const float* __restrict__ W_e, const float* __restrict__ att_e,
                                 float* __restrict__ c) {
    int t = threadIdx.x;            // 0..127
    float p = W_e[t] * att_e[t];
    #pragma unroll
    for (int off = 16; off; off >>= 1) p += __shfl_down(p, off, 32);
    if ((t & 31) == 0) c[t >> 5] = p;
}

// ---------- f32 WMMA GEMM: C[M x 128] = A[M x K] * W[K x 128] ----------
// W staged in LDS pair-interleaved: (k,n) -> lw[(k>>1)*256 + n*2 + (k&1)]
// so a lane's B fragment for K rows {k, k+1} at column n is one contiguous float2.
__global__ void gemm_wmma_f32(const float* __restrict__ A, const float* __restrict__ W,
                              float* __restrict__ C, int M, int K) {
    extern __shared__ float lw[];
    const int t = threadIdx.x;
    const int nthreads = blockDim.x;          // 128
    const int kn4 = (K * HC) >> 2;
    for (int q = t; q < kn4; q += nthreads) { // float4 reads of W
        int idx = q << 2;
        int k = idx >> 7, n = idx & 127;
        const float4 w4 = *(const float4*)(W + idx);
        float* base = &lw[((k >> 1) << 8) + (n << 1) + (k & 1)];
        base[0] = w4.x; base[2] = w4.y; base[4] = w4.z; base[6] = w4.w;
    }
    __syncthreads();

    const int wave = t >> 5, lane = t & 31;
    const int rowblock = blockIdx.x * (nthreads >> 5) + wave;
    const int row0 = rowblock * 16;
    if (row0 >= M) return;                    // after barrier: safe

    const int m     = lane & 15;              // A row within tile
    const int khalf = lane >> 4;              // 0 -> K={k0,k0+1}, 1 -> K={k0+2,k0+3}
    const int n     = lane & 15;              // B/C column within 16-wide tile

    v8f acc[8] = {};                          // 8 tiles of 16x16 -> full 128 columns
    const float* arow = A + (size_t)(row0 + m) * K + khalf * 2;

    for (int k0 = 0; k0 < K; k0 += 4) {
        v2f a = *(const v2f*)(arow + k0);     // A[m][k0+2*khalf .. +1]
        const float* bbase = &lw[((k0 >> 1) + khalf) * 256 + (n << 1)];
        v2f b[8];
        #pragma unroll
        for (int c = 0; c < 8; ++c)           // issue all LDS loads first
            b[c] = *(const v2f*)(bbase + (c << 5));
        #pragma unroll
        for (int c = 0; c < 8; ++c)
            acc[c] = __builtin_amdgcn_wmma_f32_16x16x4_f32(
                false, a, false, b[c], (short)0, acc[c], false, false);
    }

    // C/D layout: lane -> N = lane%16; VGPR r -> M = r + 8*(lane>=16)
    float* crow = C + (size_t)row0 * HC;
    const int hi = (lane >> 4) * 8;
    #pragma unroll
    for (int c = 0; c < 8; ++c)
        #pragma unroll
        for (int r = 0; r < 8; ++r)
            crow[(size_t)(r + hi) * HC + c * 16 + n] = acc[c][r];
}

// ---------- per-(node,head) attention scores ----------
__global__ void att_scores_kernel(const float* __restrict__ g,
                                  const float* __restrict__ att_src,
                                  const float* __restrict__ att_dst,
                                  float* __restrict__ a_s, float* __restrict__ a_d, int Nn) {
    int idx = blockIdx.x * blockDim.x + threadIdx.x;   // node*4 + head
    if (idx >= Nn * HEADS) return;
    int i = idx >> 2, h = idx & 3;
    const float* row = g + (size_t)i * HC + h * 32;
    const float* as  = att_src + h * 32;
    const float* ad  = att_dst + h * 32;
    float s = 0.f, d = 0.f;
    #pragma unroll
    for (int k = 0; k < 32; ++k) { float v = row[k]; s += v * as[k]; d += v * ad[k]; }
    a_s[idx] = s; a_d[idx] = d;
}

// ---------- init: out <- bias, m <- 0 (below encoded -inf), denom <- 0 ----------
__global__ void init_layer_kernel(float* __restrict__ out, const float* __restrict__ bias,
                                  unsigned* __restrict__ m_u, float* __restrict__ denom, int Nn) {
    int idx = blockIdx.x * blockDim.x + threadIdx.x;
    if (idx >= Nn * HC) return;
    out[idx] = bias[idx & 127];
    if (idx < Nn * HEADS) { m_u[idx] = 0u; denom[idx] = 0.f; }
}

// ---------- pass A: alpha = leakyrelu(a_s[src]+a_d[dst]+ea*c), segment max ----------
__global__ void edge_alpha_max_kernel(const int* __restrict__ ei, const float* __restrict__ ew,
                                      const float* __restrict__ sumptr, int E, int Nn,
                                      const float* __restrict__ a_s, const float* __restrict__ a_d,
                                      const float* __restrict__ c,
                                      float* __restrict__ alpha, unsigned* __restrict__ m_u) {
    int j = blockIdx.x * blockDim.x + threadIdx.x;
    int Et = E + Nn;
    if (j >= Et) return;
    int s, d; float ea;
    if (j < E) { s = ei[j]; d = ei[E + j]; ea = ew[j]; }
    else       { s = d = j - E; ea = sumptr[0] * (1.0f / (float)E); }
    #pragma unroll
    for (int h = 0; h < HEADS; ++h) {
        float a = a_s[s * HEADS + h] + a_d[d * HEADS + h] + ea * c[h];
        a = (a > 0.f) ? a : NEG_SLOPE * a;
        alpha[(size_t)j * HEADS + h] = a;
        atomicMax(&m_u[d * HEADS + h], enc_f32(a));
    }
}

// ---------- pass B: ex = exp(alpha - m[dst]); segment sum ----------
__global__ void edge_exp_sum_kernel(const int* __restrict__ ei, int E, int Nn,
                                    float* __restrict__ alpha, const unsigned* __restrict__ m_u,
                                    float* __restrict__ denom) {
    int j = blockIdx.x * blockDim.x + threadIdx.x;
    int Et = E + Nn;
    if (j >= Et) return;
    int d = (j < E) ? ei[E + j] : (j - E);
    #pragma unroll
    for (int h = 0; h < HEADS; ++h) {
        float m = dec_f32(m_u[d * HEADS + h]);
        float e = expf(alpha[(size_t)j * HEADS + h] - m);
        alpha[(size_t)j * HEADS + h] = e;
        atomicAdd(&denom[d * HEADS + h], e);
    }
}

// ---------- pass C: out[dst] += (ex/denom[dst]) * g[src]  (one wave per edge) ----------
__global__ void edge_aggregate_kernel(const int* __restrict__ ei, int E, int Nn,
                                      const float* __restrict__ alpha, const float* __restrict__ denom,
                                      const float* __restrict__ g, float* __restrict__ out) {
    long long gid = (long long)blockIdx.x * blockDim.x + threadIdx.x;
    int j = (int)(gid >> 5);
    int lane = (int)(gid & 31);
    int Et = E + Nn;
    if (j >= Et) return;
    int s, d;
    if (j < E) { s = ei[j]; d = ei[E + j]; }
    else       { s = d = j - E; }
    int h = lane >> 3;                                   // 4 cols/lane, 32 cols/head
    float w = alpha[(size_t)j * HEADS + h] / denom[d * HEADS + h];
    const float4 hv = *(const float4*)(g + (size_t)s * HC + lane * 4);
    float* o = out + (size_t)d * HC + lane * 4;
    atomicAdd(o + 0, w * hv.x);
    atomicAdd(o + 1, w * hv.y);
    atomicAdd(o + 2, w * hv.z);
    atomicAdd(o + 3, w * hv.w);
}

// ---------- driver ----------
extern "C" void kernel_launch(void* const* d_in, const int* in_sizes, int n_in,
                              void* d_out, int out_size, void* d_ws, size_t ws_size,
                              hipStream_t stream) {
    const float* x   = (const float*)d_in[0];
    const int*   ei  = (const int*)d_in[1];
    const float* ew  = (const float*)d_in[2];
    const float* W1  = (const float*)d_in[3];
    const float* as1 = (const float*)d_in[4];
    const float* ad1 = (const float*)d_in[5];
    const float* We1 = (const float*)d_in[6];
    const float* ae1 = (const float*)d_in[7];
    const float* b1  = (const float*)d_in[8];
    const float* W2  = (const float*)d_in[9];
    const float* as2 = (const float*)d_in[10];
    const float* ad2 = (const float*)d_in[11];
    const float* We2 = (const float*)d_in[12];
    const float* ae2 = (const float*)d_in[13];
    const float* b2  = (const float*)d_in[14];
    float* out = (float*)d_out;

    const int Nn = in_sizes[0] / 256;     // 50000
    const int E  = in_sizes[2];           // 800000
    const int Et = E + Nn;
    const int K1 = in_sizes[3] / HC;      // 256
    const int K2 = in_sizes[9] / HC;      // 128

    // workspace carve
    float*    g     = (float*)d_ws;                        // Nn*128  (GEMM output, reused L2)
    float*    h1    = g     + (size_t)Nn * HC;             // Nn*128  (layer-1 output)
    float*    a_s   = h1    + (size_t)Nn * HC;             // Nn*4
    float*    a_d   = a_s   + (size_t)Nn * HEADS;          // Nn*4
    float*    denom = a_d   + (size_t)Nn * HEADS;          // Nn*4
    unsigned* m_u   = (unsigned*)(denom + (size_t)Nn * HEADS); // Nn*4
    float*    alpha = (float*)(m_u + (size_t)Nn * HEADS);  // Et*4
    float*    csum  = alpha + (size_t)Et * HEADS;          // 1
    float*    ch    = csum + 1;                            // 4

    const int TB = 256;
    const int mblocks    = ((Nn + 15) / 16 + 3) / 4;          // 4 waves (16-row tiles) per block
    const int nh_blocks  = (Nn * HEADS + TB - 1) / TB;
    const int init_blocks= (Nn * HC + TB - 1) / TB;
    const int e_blocks   = (Et + TB - 1) / TB;
    const long long aggT = (long long)Et * 32;
    const int agg_blocks = (int)((aggT + TB - 1) / TB);

    hipMemsetAsync(csum, 0, sizeof(float), stream);
    ew_sum_kernel<<<1024, TB, 0, stream>>>(ew, csum, E);

    // ---- layer 1 ----
    gemm_wmma_f32<<<mblocks, 128, (size_t)K1 * HC * sizeof(float), stream>>>(x, W1, g, Nn, K1);
    edge_coef_kernel<<<1, 128, 0, stream>>>(We1, ae1, ch);
    att_scores_kernel<<<nh_blocks, TB, 0, stream>>>(g, as1, ad1, a_s, a_d, Nn);
    init_layer_kernel<<<init_blocks, TB, 0, stream>>>(h1, b1, m_u, denom, Nn);
    edge_alpha_max_kernel<<<e_blocks, TB, 0, stream>>>(ei, ew, csum, E, Nn, a_s, a_d, ch, alpha, m_u);
    edge_exp_sum_kernel<<<e_blocks, TB, 0, stream>>>(ei, E, Nn, alpha, m_u, denom);
    edge_aggregate_kernel<<<agg_blocks, TB, 0, stream>>>(ei, E, Nn, alpha, denom, g, h1);

    // ---- layer 2 ----
    gemm_wmma_f32<<<mblocks, 128, (size_t)K2 * HC * sizeof(float), stream>>>(h1, W2, g, Nn, K2);
    edge_coef_kernel<<<1, 128, 0, stream>>>(We2, ae2, ch);
    att_scores_kernel<<<nh_blocks, TB, 0, stream>>>(g, as2, ad2, a_s, a_d, Nn);
    init_layer_kernel<<<init_blocks, TB, 0, stream>>>(out, b2, m_u, denom, Nn);
    edge_alpha_max_kernel<<<e_blocks, TB, 0, stream>>>(ei, ew, csum, E, Nn, a_s, a_d, ch, alpha, m_u);
    edge_exp_sum_kernel<<<e_blocks, TB, 0, stream>>>(ei, E, Nn, alpha, m_u, denom);
    edge_aggregate_kernel<<<agg_blocks, TB, 0, stream>>>(ei, E, Nn, alpha, denom, g, out);

    (void)n_in; (void)out_size; (void)ws_size;
}